// SAGEConvLayer_42666205118502
// MI455X (gfx1250) — compile-verified
//
#include <hip/hip_runtime.h>

#define DIM   128
#define N_REL 3

typedef float v2f __attribute__((ext_vector_type(2)));
typedef float v8f __attribute__((ext_vector_type(8)));

// ---------------- prep: Wsum = sum_r W_self[r], bsum = sum_r b[r] ----------------
__global__ void sage_prep_kernel(const float* __restrict__ W_self,
                                 const float* __restrict__ b,
                                 float* __restrict__ Wsum,
                                 float* __restrict__ bsum) {
    int idx = blockIdx.x * blockDim.x + threadIdx.x;
    if (idx < DIM * DIM)
        Wsum[idx] = W_self[idx] + W_self[idx + DIM * DIM] + W_self[idx + 2 * DIM * DIM];
    if (idx < DIM)
        bsum[idx] = b[idx] + b[idx + DIM] + b[idx + 2 * DIM];
}

// ---------------- scatter: msg[r][dst] += h[src]; deg[r][dst] += 1 ----------------
// One wave per edge: 32 lanes x float4 = 128 floats. blockIdx.y = relation.
__global__ void __launch_bounds__(256)
sage_scatter_kernel(const float* __restrict__ h,
                    const int* __restrict__ src_base,
                    const int* __restrict__ dst_base,
                    float* __restrict__ msg_base,
                    float* __restrict__ deg_base,
                    int nE, size_t msgStride, size_t degStride) {
    const int r = blockIdx.y;
    const int wid  = (int)((blockIdx.x * blockDim.x + threadIdx.x) >> 5);
    const int lane = threadIdx.x & 31;
    if (wid >= nE) return;
    const int* src = src_base + (size_t)r * nE;
    const int* dst = dst_base + (size_t)r * nE;
    float* msg = msg_base + (size_t)r * msgStride;
    float* deg = deg_base + (size_t)r * degStride;

    const int s = src[wid];
    const int d = dst[wid];
    const float4 v = *reinterpret_cast<const float4*>(h + (size_t)s * DIM + lane * 4);
    float* mp = msg + (size_t)d * DIM + lane * 4;
    atomicAdd(mp + 0, v.x);
    atomicAdd(mp + 1, v.y);
    atomicAdd(mp + 2, v.z);
    atomicAdd(mp + 3, v.w);
    if (lane == 0) atomicAdd(deg + d, 1.0f);
}

// ---------------- shared WMMA inner machinery ----------------
// Per-wave 16-row strip x 128 cols. W staged in LDS transposed + XOR-swizzled
// (bank-conflict-free ds_load_b64 fragments within exactly 64 KB).
// Batched B loads (one dscnt wait per k-step) + A prefetched one k-step ahead.

__device__ __forceinline__ void load_W_to_lds(float* Wt, const float* __restrict__ W, int tid) {
    for (int i = tid; i < DIM * DIM; i += 256) {
        int k = i >> 7;
        int n = i & 127;
        Wt[n * DIM + (k ^ ((n & 15) << 2))] = W[i];
    }
}

__device__ __forceinline__ void gemm_phase(const float* __restrict__ Wt,
                                           const float* __restrict__ Aptr, // lane A base (+2*half)
                                           float scale, int l16, int half, int swz,
                                           v8f c[8]) {
    v2f a = *reinterpret_cast<const v2f*>(Aptr);          // k0 = 0 fragment
    for (int k0 = 0; k0 < DIM; k0 += 4) {
        const int ks = k0 + 2 * half;                      // even -> swizzle keeps pair contiguous
        v2f bf[8];
        #pragma unroll
        for (int nt = 0; nt < 8; ++nt) {
            const int n = nt * 16 + l16;
            bf[nt] = *reinterpret_cast<const v2f*>(&Wt[n * DIM + (ks ^ swz)]);
        }
        v2f as;
        as.x = a.x * scale;
        as.y = a.y * scale;
        // prefetch next A fragment (wraps to k0=0 on last iter; harmless dead load)
        a = *reinterpret_cast<const v2f*>(Aptr + ((k0 + 4) & 127));
        #pragma unroll
        for (int nt = 0; nt < 8; ++nt) {
            c[nt] = __builtin_amdgcn_wmma_f32_16x16x4_f32(
                        false, as, false, bf[nt], (short)0, c[nt], false, false);
        }
    }
}

// ---------------- fused GEMM: out = h@Wsum + sum_r (msg_r/deg_r)@Wn[r] + bsum ----
__global__ void __launch_bounds__(256)
sage_fused_gemm_kernel(const float* __restrict__ h,
                       const float* __restrict__ msg,     // [3][N*128]
                       const float* __restrict__ deg,     // [3][N]
                       const float* __restrict__ Wsum,    // [128*128]
                       const float* __restrict__ W_neigh, // [3][128*128]
                       const float* __restrict__ bsum,    // [128]
                       float* __restrict__ out, int N) {
    __shared__ float Wt[DIM * DIM];

    const int tid  = threadIdx.x;
    const int wave = tid >> 5;
    const int lane = tid & 31;
    const int half = lane >> 4;
    const int l16  = lane & 15;
    const int swz  = l16 << 2;

    const int rowBase = blockIdx.x * 128 + wave * 16;
    const int rA      = rowBase + l16;
    const int rclamp  = (rA < N) ? rA : (N - 1);

    v8f c[8] = {};

    for (int phase = 0; phase < 4; ++phase) {
        const float* W = (phase == 0) ? Wsum : (W_neigh + (size_t)(phase - 1) * DIM * DIM);
        const float* A = (phase == 0) ? h    : (msg + (size_t)(phase - 1) * N * DIM);
        float scale = 1.0f;
        if (phase > 0)
            scale = 1.0f / fmaxf(deg[(size_t)(phase - 1) * N + rclamp], 1.0f);

        __syncthreads();                 // previous phase done reading Wt
        load_W_to_lds(Wt, W, tid);
        __syncthreads();

        gemm_phase(Wt, A + (size_t)rclamp * DIM + 2 * half, scale, l16, half, swz, c);
    }

    #pragma unroll
    for (int nt = 0; nt < 8; ++nt) {
        const int col = nt * 16 + l16;
        const float bv = bsum[col];
        #pragma unroll
        for (int v = 0; v < 8; ++v) {
            const int row = rowBase + v + 8 * half;
            if (row < N)
                out[(size_t)row * DIM + col] = c[nt][v] + bv;
        }
    }
}

// ---------------- fallback: single-source GEMM (4-pass sequential path) ----------
__global__ void __launch_bounds__(256)
sage_gemm128_kernel(const float* __restrict__ A,
                    const float* __restrict__ W,
                    const float* __restrict__ bias,   // nullable
                    const float* __restrict__ deg,    // nullable
                    float* __restrict__ out,
                    int N, int accumulate) {
    __shared__ float Wt[DIM * DIM];

    const int tid  = threadIdx.x;
    load_W_to_lds(Wt, W, tid);
    __syncthreads();

    const int wave = tid >> 5;
    const int lane = tid & 31;
    const int half = lane >> 4;
    const int l16  = lane & 15;
    const int swz  = l16 << 2;

    const int rowBase = blockIdx.x * 128 + wave * 16;
    const int rA      = rowBase + l16;
    const int rclamp  = (rA < N) ? rA : (N - 1);

    float scale = 1.0f;
    if (deg) scale = 1.0f / fmaxf(deg[rclamp], 1.0f);

    v8f c[8] = {};
    gemm_phase(Wt, A + (size_t)rclamp * DIM + 2 * half, scale, l16, half, swz, c);

    #pragma unroll
    for (int nt = 0; nt < 8; ++nt) {
        const int col = nt * 16 + l16;
        const float bv = bias ? bias[col] : 0.0f;
        #pragma unroll
        for (int v = 0; v < 8; ++v) {
            const int row = rowBase + v + 8 * half;
            if (row < N) {
                const size_t o = (size_t)row * DIM + col;
                const float val = c[nt][v] + bv;
                out[o] = accumulate ? (out[o] + val) : val;
            }
        }
    }
}

extern "C" void kernel_launch(void* const* d_in, const int* in_sizes, int n_in,
                              void* d_out, int out_size, void* d_ws, size_t ws_size,
                              hipStream_t stream) {
    const float* h       = (const float*)d_in[0];
    const float* W_self  = (const float*)d_in[1];
    const float* W_neigh = (const float*)d_in[2];
    const float* b       = (const float*)d_in[3];
    const int*   e_src   = (const int*)d_in[4];
    const int*   e_dst   = (const int*)d_in[5];
    float* out = (float*)d_out;

    const int N  = in_sizes[0] / DIM;     // 50000 nodes
    const int nE = in_sizes[4] / N_REL;   // 200000 edges per relation

    const size_t fusedWords = (size_t)N_REL * N * (DIM + 1) + DIM * DIM + DIM;
    const size_t seqWords   = (size_t)N * (DIM + 1) + DIM * DIM + DIM;

    const int gemmBlocks = (N + 127) / 128;
    const int scatBlocks = (nE + 7) / 8;  // 8 waves (edges) per 256-thread block

    if (ws_size >= fusedWords * sizeof(float)) {
        // ---- fused path: out is written exactly once ----
        // ws layout: [msg 3*N*128][deg 3*N][Wsum 128*128][bsum 128]
        float* msg  = (float*)d_ws;
        float* deg  = msg + (size_t)N_REL * N * DIM;
        float* Wsum = deg + (size_t)N_REL * N;
        float* bsum = Wsum + DIM * DIM;

        sage_prep_kernel<<<(DIM * DIM + 255) / 256, 256, 0, stream>>>(W_self, b, Wsum, bsum);
        hipMemsetAsync(msg, 0, (size_t)N_REL * N * (DIM + 1) * sizeof(float), stream);

        dim3 sgrid(scatBlocks, N_REL);
        sage_scatter_kernel<<<sgrid, 256, 0, stream>>>(
            h, e_src, e_dst, msg, deg, nE, (size_t)N * DIM, (size_t)N);

        sage_fused_gemm_kernel<<<gemmBlocks, 256, 0, stream>>>(
            h, msg, deg, Wsum, W_neigh, bsum, out, N);
    } else {
        // ---- sequential fallback (smaller workspace) ----
        // ws layout: [msg N*128][deg N][Wsum 128*128][bsum 128]
        float* msg  = (float*)d_ws;
        float* deg  = msg + (size_t)N * DIM;
        float* Wsum = deg + N;
        float* bsum = Wsum + DIM * DIM;

        sage_prep_kernel<<<(DIM * DIM + 255) / 256, 256, 0, stream>>>(W_self, b, Wsum, bsum);
        sage_gemm128_kernel<<<gemmBlocks, 256, 0, stream>>>(h, Wsum, bsum, nullptr, out, N, 0);

        for (int r = 0; r < N_REL; ++r) {
            hipMemsetAsync(msg, 0, (size_t)N * (DIM + 1) * sizeof(float), stream);
            dim3 sgrid(scatBlocks, 1);
            sage_scatter_kernel<<<sgrid, 256, 0, stream>>>(
                h, e_src + (size_t)r * nE, e_dst + (size_t)r * nE, msg, deg, nE,
                (size_t)0, (size_t)0);
            sage_gemm128_kernel<<<gemmBlocks, 256, 0, stream>>>(
                msg, W_neigh + (size_t)r * DIM * DIM, nullptr, deg, out, N, 1);
        }
    }
}